// CorrelationPyramid_38671885533523
// MI455X (gfx1250) — compile-verified
//
#include <hip/hip_runtime.h>
#include <math.h>

#define BB 16
#define CC 320
#define KK 64   // 8x8 template taps

typedef float v2f __attribute__((ext_vector_type(2)));
typedef float v8f __attribute__((ext_vector_type(8)));

// ---------------------------------------------------------------------------
// Kernel 1: adaptive avg pool one template (B,C,tH,tH) -> (B,C,8,8)
// ---------------------------------------------------------------------------
__global__ void pool_kernel(const float* __restrict__ src, float* __restrict__ dst, int tH) {
    int idx = blockIdx.x * blockDim.x + threadIdx.x;      // over B*C*64
    if (idx >= BB * CC * KK) return;
    int p  = idx & 63;
    int bc = idx >> 6;
    int ky = p >> 3, kx = p & 7;
    int blk = tH >> 3;
    const float* s = src + (size_t)bc * tH * tH + (size_t)(ky * blk) * tH + kx * blk;
    float acc = 0.0f;
    for (int yy = 0; yy < blk; ++yy)
        for (int xx = 0; xx < blk; ++xx)
            acc += s[yy * tH + xx];
    dst[idx] = acc * (1.0f / (float)(blk * blk));
}

// ---------------------------------------------------------------------------
// Kernel 2: per-level fusion scores. GAP (from pooled data, mean-preserving),
// 2-layer MLP, softmax over 3 templates. One block per batch element.
// ---------------------------------------------------------------------------
__global__ void score_kernel(const float* __restrict__ pooled,   // (3,B,C,64)
                             const float* __restrict__ w1,       // (320,80)
                             const float* __restrict__ b1,       // (80)
                             const float* __restrict__ w2,       // (80,1)
                             const float* __restrict__ b2,       // (1)
                             float* __restrict__ wgt)            // (B,3)
{
    __shared__ float mean[CC];
    __shared__ float hid[80];
    __shared__ float red[128];
    __shared__ float sc[3];
    int b = blockIdx.x;
    int t = threadIdx.x;   // 128 threads

    for (int i = 0; i < 3; ++i) {
        const float* pb = pooled + (size_t)i * BB * CC * KK + (size_t)b * CC * KK;
        for (int c = t; c < CC; c += 128) {
            const float* q = pb + (size_t)c * KK;
            float s = 0.0f;
            for (int p = 0; p < KK; ++p) s += q[p];
            mean[c] = s * (1.0f / 64.0f);
        }
        __syncthreads();
        if (t < 80) {
            float h = b1[t];
            for (int c = 0; c < CC; ++c) h += mean[c] * w1[c * 80 + t];
            hid[t] = fmaxf(h, 0.0f);
        }
        __syncthreads();
        red[t] = (t < 80) ? hid[t] * w2[t] : 0.0f;
        __syncthreads();
        for (int s = 64; s > 0; s >>= 1) {
            if (t < s) red[t] += red[t + s];
            __syncthreads();
        }
        if (t == 0) sc[i] = red[0] + b2[0];
        __syncthreads();
    }
    if (t == 0) {
        float m  = fmaxf(sc[0], fmaxf(sc[1], sc[2]));
        float e0 = __expf(sc[0] - m), e1 = __expf(sc[1] - m), e2 = __expf(sc[2] - m);
        float inv = 1.0f / (e0 + e1 + e2);
        wgt[b * 3 + 0] = e0 * inv;
        wgt[b * 3 + 1] = e1 * inv;
        wgt[b * 3 + 2] = e2 * inv;
    }
}

// ---------------------------------------------------------------------------
// Kernel 3: fuse pooled templates with softmax weights, L2-normalize over C.
// Grid: B*64 blocks (one per (b, tap)), 320 threads (one per channel).
// Output layout is fragment-ready / async-copy friendly:
//   tnI[b][c/2][tap][2] = { tn[c][tap], tn[c+1][tap] }
// so one ds_load_b64 yields a complete B-fragment v2f for a K-pair.
// ---------------------------------------------------------------------------
__global__ void fuse_norm_kernel(const float* __restrict__ pooled,  // (3,B,C,64)
                                 const float* __restrict__ wgt,     // (B,3)
                                 float* __restrict__ tn)            // (B,C/2,64,2)
{
    __shared__ float red[CC];
    __shared__ float inv;
    int p = blockIdx.x & 63;
    int b = blockIdx.x >> 6;
    int c = threadIdx.x;

    float a0 = wgt[b * 3 + 0], a1 = wgt[b * 3 + 1], a2 = wgt[b * 3 + 2];
    size_t off = (size_t)b * CC * KK + (size_t)c * KK + p;
    const size_t TSZ = (size_t)BB * CC * KK;
    float f = a0 * pooled[off] + a1 * pooled[TSZ + off] + a2 * pooled[2 * TSZ + off];

    red[c] = f * f;
    __syncthreads();
    for (int s = 256; s > 0; s >>= 1) {
        if (c < s && c + s < CC) red[c] += red[c + s];
        __syncthreads();
    }
    if (c == 0) inv = 1.0f / fmaxf(sqrtf(red[0]), 1e-12f);
    __syncthreads();
    // interleaved pair layout
    size_t oi = (size_t)b * CC * KK + (size_t)(c >> 1) * 128 + (size_t)p * 2 + (c & 1);
    tn[oi] = f * inv;
}

// ---------------------------------------------------------------------------
// Kernel 4: reciprocal channel-L2-norm map of the search tensor: (B,H,W)
// ---------------------------------------------------------------------------
__global__ void rnorm_kernel(const float* __restrict__ S, float* __restrict__ rn, int HW) {
    int idx = blockIdx.x * blockDim.x + threadIdx.x;
    if (idx >= BB * HW) return;
    int b = idx / HW, p = idx % HW;
    const float* s = S + (size_t)b * CC * HW + p;
    float ss = 0.0f;
    for (int c = 0; c < CC; ++c) {
        float v = s[(size_t)c * HW];
        ss += v * v;
    }
    rn[idx] = 1.0f / fmaxf(sqrtf(ss), 1e-12f);
}

// ---------------------------------------------------------------------------
// Kernel 5: WMMA GEMM.  Per batch b:
//   D[m, n] = rn[m] * sum_c  S[c, m] * T_norm[c, n]   (m: HW pixels, n: 64 taps)
//
// - Template (B-matrix, 80 KB/batch) staged into LDS with the CDNA5 async
//   copy path (GLOBAL_LOAD_ASYNC_TO_LDS_B128 + s_wait_asynccnt) in two 40 KB
//   half-K chunks; shared by all 4 waves and all K-steps of the block.
// - Each wave owns TWO 16-pixel M-tiles -> B-fragments reused, 8 WMMAs per
//   K-step against 4 NT global A-loads + ds_load_2addr_b64 B-loads.
// - A stream is software-pipelined (double-buffered) so its global-load
//   latency overlaps the WMMA block instead of stalling it.
// - Search L2-normalization is linear in A -> folded into the accumulator
//   scale at store time (removes the per-K-step v_mul chain).
// Block = 128 threads (4 waves); grid = (HW/128, B).  No divergence: EXEC=all-1s.
// ---------------------------------------------------------------------------
__global__ void corr_wmma_kernel(const float* __restrict__ S,   // (B,C,HW)
                                 const float* __restrict__ tn,  // (B,C/2,64,2)
                                 const float* __restrict__ rn,  // (B,HW)
                                 float* __restrict__ D,         // (B,HW,64)
                                 int HW)
{
    __shared__ __align__(16) float Bls[CC * KK / 2];   // 40 KB: half of the K range

    int b    = blockIdx.y;
    int wave = threadIdx.x >> 5;
    int lane = threadIdx.x & 31;
    int half = lane >> 4;        // 0: K=c0,c0+1   1: K=c0+2,c0+3
    int l    = lane & 15;
    int m0   = (blockIdx.x * 4 + wave) * 32;           // two M-tiles per wave

    const float* Sb = S  + (size_t)b * CC * HW;
    const float* tb = tn + (size_t)b * CC * KK;        // interleaved pair layout
    const float* rb = rn + (size_t)b * HW;
    float*       Db = D  + (size_t)b * HW * KK;

    const int mA = m0 + l;
    const int mB = m0 + 16 + l;

    const v2f* BshV = (const v2f*)Bls;
    unsigned int lbase = (unsigned int)(size_t)&Bls[0];   // low 32 bits = LDS offset

    v8f accA0 = {}, accA1 = {}, accA2 = {}, accA3 = {};
    v8f accB0 = {}, accB1 = {}, accB2 = {}, accB3 = {};

    // raw A-fragment loads (no scaling; rnorm folded into accumulator store)
    auto loadA = [&](int c, v2f& xA, v2f& xB) {
        int ca = c + 2 * half;
        xA.x = __builtin_nontemporal_load(&Sb[(size_t)ca * HW + mA]);
        xA.y = __builtin_nontemporal_load(&Sb[(size_t)(ca + 1) * HW + mA]);
        xB.x = __builtin_nontemporal_load(&Sb[(size_t)ca * HW + mB]);
        xB.y = __builtin_nontemporal_load(&Sb[(size_t)(ca + 1) * HW + mB]);
    };

    for (int kh = 0; kh < 2; ++kh) {
        // ---- async-stage 40 KB of the template for channels [kh*160, kh*160+160)
        {
            unsigned int loff = lbase + (unsigned int)threadIdx.x * 16u;
            const float* gp   = tb + (size_t)kh * (CC * KK / 2) + (size_t)threadIdx.x * 4;
            for (int it = 0; it < (CC * KK / 2) / (128 * 4); ++it) {   // 20 iters
                asm volatile("global_load_async_to_lds_b128 %0, %1, off"
                             :: "v"(loff), "v"(gp) : "memory");
                loff += 128u * 16u;
                gp   += 128 * 4;
            }
            asm volatile("s_wait_asynccnt 0" ::: "memory");
            __syncthreads();
        }

        const int cs = kh * (CC / 2);
        const int ce = cs + (CC / 2);

        v2f aAc, aBc;
        loadA(cs, aAc, aBc);                       // prologue: first A fragment

        for (int c0 = cs; c0 < ce; c0 += 4) {
            // issue NEXT iteration's A loads now (clamped at the tail);
            // their wait lands after this iteration's WMMA block
            int cn = (c0 + 4 < ce) ? (c0 + 4) : c0;
            v2f aAn, aBn;
            loadA(cn, aAn, aBn);

            // prefetch the A stream further ahead into L2 (global_prefetch_b8)
            int cpf = (c0 + 2 * half + 16 < CC) ? (c0 + 2 * half + 16) : (c0 + 2 * half);
            __builtin_prefetch(&Sb[(size_t)cpf * HW + mA], 0, 2);

            // B fragments from LDS (pairs merge into ds_load_2addr_b64)
            int cpl = (c0 - cs) / 2 + half;
            v2f b0 = BshV[(size_t)cpl * KK + 0  + l];
            v2f b1 = BshV[(size_t)cpl * KK + 16 + l];
            v2f b2 = BshV[(size_t)cpl * KK + 32 + l];
            v2f b3 = BshV[(size_t)cpl * KK + 48 + l];

            accA0 = __builtin_amdgcn_wmma_f32_16x16x4_f32(false, aAc, false, b0, (short)0, accA0, false, false);
            accA1 = __builtin_amdgcn_wmma_f32_16x16x4_f32(false, aAc, false, b1, (short)0, accA1, false, false);
            accA2 = __builtin_amdgcn_wmma_f32_16x16x4_f32(false, aAc, false, b2, (short)0, accA2, false, false);
            accA3 = __builtin_amdgcn_wmma_f32_16x16x4_f32(false, aAc, false, b3, (short)0, accA3, false, false);
            accB0 = __builtin_amdgcn_wmma_f32_16x16x4_f32(false, aBc, false, b0, (short)0, accB0, false, false);
            accB1 = __builtin_amdgcn_wmma_f32_16x16x4_f32(false, aBc, false, b1, (short)0, accB1, false, false);
            accB2 = __builtin_amdgcn_wmma_f32_16x16x4_f32(false, aBc, false, b2, (short)0, accB2, false, false);
            accB3 = __builtin_amdgcn_wmma_f32_16x16x4_f32(false, aBc, false, b3, (short)0, accB3, false, false);

            aAc = aAn;
            aBc = aBn;
        }
        __syncthreads();   // before restaging overwrites the LDS tile
    }

    // C/D layout: VGPR r, lane L -> M = r + 8*(L>=16), N = L&15.
    // Apply the deferred search-pixel inverse norm (row scale of D) here.
#pragma unroll
    for (int r = 0; r < 8; ++r) {
        int moA = m0 + r + 8 * half;
        float rvA = rb[moA];
        float rvB = rb[moA + 16];
        float* dA = Db + (size_t)moA * KK + l;
        dA[0]  = accA0[r] * rvA;
        dA[16] = accA1[r] * rvA;
        dA[32] = accA2[r] * rvA;
        dA[48] = accA3[r] * rvA;
        float* dB = dA + (size_t)16 * KK;
        dB[0]  = accB0[r] * rvB;
        dB[16] = accB1[r] * rvB;
        dB[32] = accB2[r] * rvB;
        dB[48] = accB3[r] * rvB;
    }
}

// ---------------------------------------------------------------------------
// Kernel 6: shift-and-add gather:  out[b,y,x] = sum_{i,j} D[b, (y+i)W+(x+j), 8i+j]
// ---------------------------------------------------------------------------
__global__ void gather_kernel(const float* __restrict__ D, float* __restrict__ out,
                              int W, int Ho) {
    int per = Ho * Ho;
    int idx = blockIdx.x * blockDim.x + threadIdx.x;
    if (idx >= BB * per) return;
    int b = idx / per, q = idx % per;
    int y = q / Ho, x = q % Ho;
    const float* Db = D + (size_t)b * W * W * KK;
    float acc = 0.0f;
#pragma unroll
    for (int i = 0; i < 8; ++i) {
        const float* row = Db + ((size_t)(y + i) * W + x) * KK + i * 8;
#pragma unroll
        for (int j = 0; j < 8; ++j)
            acc += row[(size_t)j * KK + j];
    }
    out[idx] = acc;
}

// ---------------------------------------------------------------------------
extern "C" void kernel_launch(void* const* d_in, const int* in_sizes, int n_in,
                              void* d_out, int out_size, void* d_ws, size_t ws_size,
                              hipStream_t stream) {
    (void)in_sizes; (void)n_in; (void)out_size; (void)ws_size;

    // input order: t{0,1,2}_p{3,4,5}, s_p{3,4,5}, fp{3,4,5}_{w1,b1,w2,b2}
    const float* T[3][3];   // [level][template]
    for (int L = 0; L < 3; ++L)
        for (int i = 0; i < 3; ++i)
            T[L][i] = (const float*)d_in[i * 3 + L];
    const float* Sarr[3] = { (const float*)d_in[9], (const float*)d_in[10], (const float*)d_in[11] };
    const float *W1[3], *B1[3], *W2[3], *B2[3];
    for (int L = 0; L < 3; ++L) {
        W1[L] = (const float*)d_in[12 + L * 4];
        B1[L] = (const float*)d_in[13 + L * 4];
        W2[L] = (const float*)d_in[14 + L * 4];
        B2[L] = (const float*)d_in[15 + L * 4];
    }

    const int tHs[3] = { 32, 16, 8 };
    const int sHs[3] = { 64, 32, 16 };
    const int HWs[3] = { 4096, 1024, 256 };
    const int Hos[3] = { 57, 25, 9 };

    // workspace layout (float elements)
    const size_t PSZ  = (size_t)BB * CC * KK;                      // 327680
    const size_t WGT  = 9 * PSZ;                                   // pooled: 9 tensors
    const size_t TNO  = WGT + 3 * (size_t)BB * 3;                  // weights: 3*(B*3)
    const size_t RN0  = TNO + 3 * PSZ;                             // tn: 3 tensors
    const size_t RNoff[3] = { RN0,
                              RN0 + (size_t)BB * 4096,
                              RN0 + (size_t)BB * 4096 + (size_t)BB * 1024 };
    const size_t DD   = RN0 + (size_t)BB * (4096 + 1024 + 256);
    const size_t Doff[3] = { DD,
                             DD + (size_t)BB * 4096 * KK,
                             DD + (size_t)BB * 4096 * KK + (size_t)BB * 1024 * KK };
    const size_t Ooff[3] = { 0,
                             (size_t)BB * 57 * 57,
                             (size_t)BB * 57 * 57 + (size_t)BB * 25 * 25 };

    float* ws  = (float*)d_ws;
    float* out = (float*)d_out;

    // 1) pool all 9 templates to 8x8 (reads full-res templates exactly once)
    for (int L = 0; L < 3; ++L)
        for (int i = 0; i < 3; ++i)
            pool_kernel<<<(int)((PSZ + 255) / 256), 256, 0, stream>>>(
                T[L][i], ws + (size_t)(L * 3 + i) * PSZ, tHs[L]);

    // 2) fusion scores + softmax weights per level
    for (int L = 0; L < 3; ++L)
        score_kernel<<<BB, 128, 0, stream>>>(
            ws + (size_t)(L * 3) * PSZ, W1[L], B1[L], W2[L], B2[L],
            ws + WGT + (size_t)L * BB * 3);

    // 3) fuse + L2-normalize template (interleaved B-matrix layout)
    for (int L = 0; L < 3; ++L)
        fuse_norm_kernel<<<BB * KK, CC, 0, stream>>>(
            ws + (size_t)(L * 3) * PSZ, ws + WGT + (size_t)L * BB * 3,
            ws + TNO + (size_t)L * PSZ);

    // 4) search reciprocal-norm maps
    for (int L = 0; L < 3; ++L)
        rnorm_kernel<<<(BB * HWs[L] + 255) / 256, 256, 0, stream>>>(
            Sarr[L], ws + RNoff[L], HWs[L]);

    // 5) WMMA correlation GEMM  D = diag(rn) * S^T * T_norm (async-LDS staged B)
    for (int L = 0; L < 3; ++L) {
        dim3 grid(HWs[L] / 128, BB);
        corr_wmma_kernel<<<grid, 128, 0, stream>>>(
            Sarr[L], ws + TNO + (size_t)L * PSZ, ws + RNoff[L],
            ws + Doff[L], HWs[L]);
    }

    // 6) shift-and-add gather -> outputs
    for (int L = 0; L < 3; ++L) {
        int n = BB * Hos[L] * Hos[L];
        gather_kernel<<<(n + 255) / 256, 256, 0, stream>>>(
            ws + Doff[L], out + Ooff[L], sHs[L], Hos[L]);
    }
}